// NAA_86199993631438
// MI455X (gfx1250) — compile-verified
//
#include <hip/hip_runtime.h>
#include <stdint.h>

typedef __attribute__((ext_vector_type(2))) float v2f;
typedef __attribute__((ext_vector_type(4))) float v4f;
typedef __attribute__((ext_vector_type(8))) float v8f;

#define N_CLASSES 6000
#define ATT       1024
#define GM1       14      // groups actually written (k=1..14 in torch loop)
#define GSIZE     64
#define N_UNSEEN  1200
#define N_SEEN    4800

// ---------------------------------------------------------------------------
// Workspace layout (4-byte units):
//   [0      , 1024)   maskBits[1024]  (u32; bit l set => column replaced in row l)
//   [1024   , 1040)   cnt[16]         (int; distinct replaced columns per row)
//   [1040   , 1056)   rep[16]         (float; replacement value per row l)
//   [1056   , 1072)   rep2[16]        (float; rep^2)
//   [1088   , 17472)  Bmat[1024][16]  (f32; col0=1, col1=0, col l = mask)
//   [17472  , 113472) inv[6000][16]   (f32; 1/max(||m[c,l]||,1e-12))
// Total ~454 KB.
// ---------------------------------------------------------------------------

__global__ void naa_build(const float* __restrict__ betas,
                          const int*   __restrict__ gcols,
                          uint32_t*    __restrict__ maskBits,
                          int*         __restrict__ cnt,
                          float*       __restrict__ rep,
                          float*       __restrict__ rep2,
                          float*       __restrict__ Bmat) {
  const int tid = threadIdx.x;                 // 1024 threads, single block
  maskBits[tid] = 0u;
  if (tid < 16) {
    cnt[tid] = 0;
    float b = (tid >= 2) ? betas[tid - 2] : 0.0f;
    rep[tid]  = b;
    rep2[tid] = b * b;
  }
  __syncthreads();
  if (tid < GM1 * GSIZE) {                     // scatter: distinct-set semantics
    int k   = tid / GSIZE;
    int col = gcols[tid];
    atomicOr(&maskBits[col], 1u << (k + 2));
  }
  __syncthreads();
  uint32_t bits = maskBits[tid];
  Bmat[tid * 16 + 0] = 1.0f;                   // col 0 -> base sum of squares
  Bmat[tid * 16 + 1] = 0.0f;                   // col 1 unused (row1 == row0)
#pragma unroll
  for (int l = 2; l < 16; ++l) {
    int m = (int)((bits >> l) & 1u);
    Bmat[tid * 16 + l] = (float)m;
    if (m) atomicAdd(&cnt[l], 1);
  }
}

// One full wave (32 lanes) per 16 classes.  P = X2 * B via V_WMMA_F32_16X16X4_F32.
__global__ void naa_norms(const float* __restrict__ attr,
                          const float* __restrict__ Bmat,
                          const int*   __restrict__ cnt,
                          const float* __restrict__ rep2,
                          float*       __restrict__ inv) {
  const int lane  = threadIdx.x;               // 0..31, EXEC all ones for WMMA
  const int cbase = blockIdx.x * 16;
  const int half  = lane >> 4;                 // K sub-block (ISA A/B layout)
  const int m     = lane & 15;                 // M (A) / N (B) index

  v8f acc = {};
  const float* arow = attr + (size_t)(cbase + m) * ATT;
  for (int k0 = 0; k0 < ATT; k0 += 4) {
    const int kA = k0 + half * 2;
    float a0 = arow[kA];
    float a1 = arow[kA + 1];
    v2f Af = {a0 * a0, a1 * a1};               // A[M][K], squared on the fly
    v2f Bf = {Bmat[kA * 16 + m], Bmat[(kA + 1) * 16 + m]};  // B[K][N]
    acc = __builtin_amdgcn_wmma_f32_16x16x4_f32(
        /*neg_a=*/false, Af, /*neg_b=*/false, Bf,
        /*c_mod=*/(short)0, acc, /*reuse_a=*/false, /*reuse_b=*/false);
  }

  // D layout: vgpr r, lane L -> C[r + 8*(L/16)][L%16]
  __shared__ float P[16 * 16];
#pragma unroll
  for (int r = 0; r < 8; ++r) P[(r + 8 * half) * 16 + m] = acc[r];
  __syncthreads();

  for (int idx = lane; idx < 256; idx += 32) {
    int cl = idx >> 4, l = idx & 15;
    float S = P[cl * 16 + 0];                  // base sum of squares
    float v = (l < 2) ? S
                      : (S - P[cl * 16 + l] + (float)cnt[l] * rep2[l]);
    v = fmaxf(v, 0.0f);
    inv[(size_t)(cbase + cl) * 16 + l] = 1.0f / fmaxf(sqrtf(v), 1e-12f);
  }
}

// out[a*W + u*16 + l] = (replaced(l,a) ? rep[l] : attr[cls[u],a]) * inv[cls[u],l]
// One v4f along j per thread (4 | 16 => never crosses a class boundary).
__global__ void naa_write(const float*    __restrict__ attr,
                          const uint32_t* __restrict__ maskBits,
                          const float*    __restrict__ inv,
                          const float*    __restrict__ rep,
                          const int*      __restrict__ cls,   // null => identity
                          int W,
                          float*          __restrict__ out) {
  const int    w4     = W >> 2;
  const size_t total4 = (size_t)ATT * (size_t)w4;
  size_t idx4 = (size_t)blockIdx.x * blockDim.x + threadIdx.x;
  if (idx4 >= total4) return;

  const int a  = (int)(idx4 / (size_t)w4);
  const int j  = (int)(idx4 % (size_t)w4) * 4;
  const int u  = j >> 4;
  const int l0 = j & 15;                       // multiple of 4
  const int c  = cls ? cls[u] : u;

  const float    x    = attr[(size_t)c * ATT + a];      // L2-resident gather
  const uint32_t bits = maskBits[a];
  const v4f      iv   = *(const v4f*)(inv + (size_t)c * 16 + l0);
  const v4f      rp   = *(const v4f*)(rep + l0);

  v4f o;
  o.x = (((bits >> (l0 + 0)) & 1u) ? rp.x : x) * iv.x;
  o.y = (((bits >> (l0 + 1)) & 1u) ? rp.y : x) * iv.y;
  o.z = (((bits >> (l0 + 2)) & 1u) ? rp.z : x) * iv.z;
  o.w = (((bits >> (l0 + 3)) & 1u) ? rp.w : x) * iv.w;

  __builtin_nontemporal_store(o, (v4f*)(out + (size_t)a * W + j));
}

extern "C" void kernel_launch(void* const* d_in, const int* in_sizes, int n_in,
                              void* d_out, int out_size, void* d_ws, size_t ws_size,
                              hipStream_t stream) {
  const float* attr   = (const float*)d_in[0];   // [6000,1024] f32
  const float* betas  = (const float*)d_in[1];   // [1,15] f32
  const int*   gcols  = (const int*)  d_in[2];   // [14,64] i32
  const int*   unseen = (const int*)  d_in[3];   // [1200] i32
  const int*   seen   = (const int*)  d_in[4];   // [4800] i32
  float*       out    = (float*)d_out;

  uint32_t* ws32     = (uint32_t*)d_ws;
  uint32_t* maskBits = ws32;
  int*      cnt      = (int*)  (ws32 + 1024);
  float*    rep      = (float*)(ws32 + 1040);
  float*    rep2     = (float*)(ws32 + 1056);
  float*    Bmat     = (float*)(ws32 + 1088);
  float*    inv      = (float*)(ws32 + 1088 + ATT * 16);

  naa_build<<<1, 1024, 0, stream>>>(betas, gcols, maskBits, cnt, rep, rep2, Bmat);
  naa_norms<<<N_CLASSES / 16, 32, 0, stream>>>(attr, Bmat, cnt, rep2, inv);

  const int Wz = N_UNSEEN * 16, Ws = N_SEEN * 16, Wg = N_CLASSES * 16;
  auto nb = [](size_t total4) { return (unsigned)((total4 + 255) / 256); };

  naa_write<<<nb((size_t)ATT * (Wz / 4)), 256, 0, stream>>>(
      attr, maskBits, inv, rep, unseen, Wz, out);
  naa_write<<<nb((size_t)ATT * (Ws / 4)), 256, 0, stream>>>(
      attr, maskBits, inv, rep, seen, Ws, out + (size_t)ATT * Wz);
  naa_write<<<nb((size_t)ATT * (Wg / 4)), 256, 0, stream>>>(
      attr, maskBits, inv, rep, (const int*)nullptr, Wg,
      out + (size_t)ATT * (Wz + Ws));
}